// MMDLoss_18313740550844
// MI455X (gfx1250) — compile-verified
//
#include <hip/hip_runtime.h>
#include <hip/hip_bf16.h>

typedef __attribute__((ext_vector_type(2))) float v2f;
typedef __attribute__((ext_vector_type(8))) float v8f;

#define BB 2048
#define TT 128
#define DD 256
#define M1 4096   // B * samples
#define M2 8192   // 2 * M1

// ---------------------------------------------------------------- gather
// total[m][:] for m<M1 comes from source[b=m/2][indices[m%2]][:],
// for m>=M1 from target. One block (64 threads) copies one 1KB row as float4.
__global__ void mmd_gather(const float* __restrict__ src,
                           const float* __restrict__ tgt,
                           const int* __restrict__ idx,
                           float* __restrict__ total) {
    int m = blockIdx.x;
    int lane = threadIdx.x;            // 0..63, one float4 each
    bool isT = m >= M1;
    int mm = isT ? m - M1 : m;
    int b = mm >> 1;
    int s = mm & 1;
    int t = idx[s];
    const float* row = (isT ? tgt : src) + ((size_t)b * TT + t) * DD;
    float4 v = ((const float4*)row)[lane];
    ((float4*)(total + (size_t)m * DD))[lane] = v;
}

// ---------------------------------------------------------------- row norms
__global__ void mmd_sq(const float* __restrict__ total, float* __restrict__ sq) {
    int m = blockIdx.x * blockDim.x + threadIdx.x;   // 8192 threads
    const float4* row = (const float4*)(total + (size_t)m * DD);
    float acc = 0.f;
#pragma unroll 8
    for (int i = 0; i < DD / 4; ++i) {
        float4 v = row[i];
        acc += v.x * v.x + v.y * v.y + v.z * v.z + v.w * v.w;
    }
    sq[m] = acc;
}

// ---------------------------------------------------------------- zero accum
__global__ void mmd_zero(double* acc) {
    if (threadIdx.x < 8) acc[threadIdx.x] = 0.0;
}

// ---------------------------------------------------------------- bandwidth
__global__ void mmd_bw(double* acc) {
    // bandwidth = sum(L2) / (M2^2 - M2) / KERNEL_MUL^(KERNEL_NUM/2)
    double denom = (double)M2 * (double)M2 - (double)M2;
    acc[1] = acc[0] / denom / 4.0;     // 2.0^(5//2) = 4
}

// ---------------------------------------------------------------- Gram passes
// PASS 1: accumulate sum of clamped L2 into acc[0].
// PASS 2: accumulate 5-kernel sums per quadrant into acc[2..5] (XX,YY,XY,YX).
// Wave -> one 16x16 tile via V_WMMA_F32_16X16X4_F32; block = 8 waves = 64x32.
template <int PASS>
__global__ void mmd_gram(const float* __restrict__ total,
                         const float* __restrict__ sq,
                         double* __restrict__ acc) {
    const int tid  = threadIdx.x;
    const int lane = tid & 31;
    const int w    = tid >> 5;          // wave id 0..7
    const int half = lane >> 4;         // 0: K,K+1   1: K+2,K+3
    const int lrow = lane & 15;

    const int m0 = blockIdx.y * 64 + (w & 3) * 16;
    const int n0 = blockIdx.x * 32 + (w >> 2) * 16;

    const float* Arow = total + (size_t)(m0 + lrow) * DD + 2 * half;
    const float* Brow = total + (size_t)(n0 + lrow) * DD + 2 * half;

    v8f c0 = {}, c1 = {};
#pragma unroll 4
    for (int k = 0; k < DD; k += 8) {
        v2f a0 = *(const v2f*)(Arow + k);
        v2f b0 = *(const v2f*)(Brow + k);
        c0 = __builtin_amdgcn_wmma_f32_16x16x4_f32(false, a0, false, b0,
                                                   (short)0, c0, false, false);
        v2f a1 = *(const v2f*)(Arow + k + 4);
        v2f b1 = *(const v2f*)(Brow + k + 4);
        c1 = __builtin_amdgcn_wmma_f32_16x16x4_f32(false, a1, false, b1,
                                                   (short)0, c1, false, false);
    }
    v8f c = c0 + c1;   // G[m0 + v + 8*half][n0 + lrow] in c[v]

    float sqn = sq[n0 + lrow];
    float sqm[8];
#pragma unroll
    for (int v = 0; v < 8; ++v) sqm[v] = sq[m0 + 8 * half + v];

    float inv16bw = 0.f;
    if (PASS == 2) inv16bw = (float)(1.0 / (16.0 * acc[1]));

    double part = 0.0;
#pragma unroll
    for (int v = 0; v < 8; ++v) {
        float L2 = fmaxf(sqm[v] + sqn - 2.0f * c[v], 0.0f);
        if (PASS == 1) {
            part += (double)L2;
        } else {
            // sum_i exp(-L2/(bw*2^i)), i=0..4  ==  t^16+t^8+t^4+t^2+t,
            // t = exp(-L2/(16*bw)):  one v_exp_f32 + 4 squarings.
            float t   = __expf(-L2 * inv16bw);
            float t2  = t * t;
            float t4  = t2 * t2;
            float t8  = t4 * t4;
            float t16 = t8 * t8;
            part += (double)(t + t2 + t4 + t8 + t16);
        }
    }

    __shared__ double red[256];
    red[tid] = part;
    __syncthreads();
#pragma unroll
    for (int s2 = 128; s2 > 0; s2 >>= 1) {
        if (tid < s2) red[tid] += red[tid + s2];
        __syncthreads();
    }
    if (tid == 0) {
        int slot;
        if (PASS == 1) {
            slot = 0;
        } else {
            int qr = (m0 >= M1), qc = (n0 >= M1);
            // XX=2 (0,0)  YY=3 (1,1)  XY=4 (0,1)  YX=5 (1,0)
            slot = 2 + (qr ? (qc ? 1 : 3) : (qc ? 2 : 0));
        }
        atomicAdd(acc + slot, red[0]);   // global_atomic_add_f64
    }
}

// ---------------------------------------------------------------- finalize
__global__ void mmd_final(const double* __restrict__ acc, float* __restrict__ out) {
    double denom = (double)M1 * (double)M1;   // each block mean over 4096^2
    out[0] = (float)((acc[2] + acc[3] - acc[4] - acc[5]) / denom);
}

extern "C" void kernel_launch(void* const* d_in, const int* in_sizes, int n_in,
                              void* d_out, int out_size, void* d_ws, size_t ws_size,
                              hipStream_t stream) {
    const float* src = (const float*)d_in[0];
    const float* tgt = (const float*)d_in[1];
    const int*   idx = (const int*)d_in[2];
    float* out = (float*)d_out;

    char* ws = (char*)d_ws;
    float*  total = (float*)ws;                                    // 8 MB
    float*  sq    = (float*)(ws + (size_t)M2 * DD * 4);            // 32 KB
    double* acc   = (double*)(ws + (size_t)M2 * DD * 4 + M2 * 4);  // 64 B

    mmd_zero<<<1, 32, 0, stream>>>(acc);
    mmd_gather<<<M2, 64, 0, stream>>>(src, tgt, idx, total);
    mmd_sq<<<M2 / 256, 256, 0, stream>>>(total, sq);

    dim3 grid(M2 / 32, M2 / 64), block(256);
    mmd_gram<1><<<grid, block, 0, stream>>>(total, sq, acc);
    mmd_bw<<<1, 1, 0, stream>>>(acc);
    mmd_gram<2><<<grid, block, 0, stream>>>(total, sq, acc);
    mmd_final<<<1, 1, 0, stream>>>(acc, out);
}